// GCN_42374147342368
// MI455X (gfx1250) — compile-verified
//
#include <hip/hip_runtime.h>

typedef __attribute__((ext_vector_type(2))) float v2f;
typedef __attribute__((ext_vector_type(8))) float v8f;

#define F_IN 512
#define HID  16
#define OUTF 2

// ---------------- degree / norm ----------------
__global__ void k_deg_init(float* __restrict__ deg, int N) {
  int n = blockIdx.x * blockDim.x + threadIdx.x;
  if (n < N) deg[n] = 1.0f;                       // self-loop contributes 1
}

__global__ void k_deg_count(const int* __restrict__ dst, float* __restrict__ deg, int E) {
  int e = blockIdx.x * blockDim.x + threadIdx.x;
  if (e < E) atomicAdd(&deg[dst[e]], 1.0f);
}

__global__ void k_rsqrt(float* __restrict__ deg, int N) {
  int n = blockIdx.x * blockDim.x + threadIdx.x;
  if (n < N) deg[n] = rsqrtf(deg[n]);             // in place: deg -> dinv
}

// ---------------- layer-1 GEMM: h = x @ W1 via V_WMMA_F32_16X16X4_F32 ----------------
// One wave per 16-node tile. A = 16x4 fp32 (2 contiguous floats/lane),
// B = 4x16 from W1, C/D = 16x16 fp32 accumulator (8 VGPRs).
__global__ __launch_bounds__(256) void k_gemm1_wmma(
    const float* __restrict__ x, const float* __restrict__ W1,
    float* __restrict__ h, int N, int nTiles) {
  int wave = blockIdx.x * (blockDim.x >> 5) + (threadIdx.x >> 5);
  if (wave >= nTiles) return;                     // whole-wave uniform exit (EXEC all-1 for WMMA)
  int lane = threadIdx.x & 31;
  int col  = lane & 15;                           // M for A-frag, N for B/C/D frags
  int hi   = lane >> 4;                           // selects K pair {0,1} vs {2,3}
  int tileBase = wave * 16;

  int arow = tileBase + col;                      // node this lane loads for A
  if (arow > N - 1) arow = N - 1;                 // clamp (keeps EXEC uniform)
  const float* xp = x  + (size_t)arow * F_IN + hi * 2;
  const float* wp = W1 + (size_t)(hi * 2) * HID + col;

  v8f acc = {};
#pragma unroll 4
  for (int kb = 0; kb < F_IN; kb += 4) {
    // A 16x4: lanes 0-15 hold K=kb+0,kb+1 ; lanes 16-31 hold K=kb+2,kb+3 (row M = lane&15)
    v2f a = __builtin_nontemporal_load((const v2f*)(xp + kb));   // stream x, don't pollute L2
    // B 4x16: K = kb + 2*hi + vgpr, N = lane&15
    v2f b;
    b.x = wp[kb * HID];
    b.y = wp[kb * HID + HID];
    acc = __builtin_amdgcn_wmma_f32_16x16x4_f32(
        /*neg_a=*/false, a, /*neg_b=*/false, b,
        /*c_mod=*/(short)0, acc, /*reuse_a=*/false, /*reuse_b=*/false);
  }
  // D 16x16: VGPR r, lanes 0-15 -> M=r, lanes 16-31 -> M=r+8 ; N = lane&15
  if (tileBase + 16 <= N) {
    // full tile (always taken when N % 16 == 0): unguarded, immediate-offset stores
    float* outp = h + ((size_t)(tileBase + 8 * hi)) * HID + col;
#pragma unroll
    for (int r = 0; r < 8; ++r) outp[r * HID] = acc[r];
  } else {
    // ragged last tile
#pragma unroll
    for (int r = 0; r < 8; ++r) {
      int m = tileBase + r + 8 * hi;
      if (m < N) h[(size_t)m * HID + col] = acc[r];
    }
  }
}

// ---------------- layer-1 aggregation ----------------
// acc1 = h * dinv^2 (self-loop) + b1   (bias folded into init; sum is linear)
__global__ void k_agg1_init(const float* __restrict__ h, const float* __restrict__ dinv,
                            const float* __restrict__ b1, float* __restrict__ acc1, int N) {
  int t = blockIdx.x * blockDim.x + threadIdx.x;
  if (t >= N * HID) return;
  int n = t >> 4, j = t & 15;
  float dv = dinv[n];
  acc1[t] = h[t] * dv * dv + b1[j];
}

// 4 lanes per edge; each lane reads a float4 of the message (global_load_b128 from
// L2-resident h) and issues 4 fp32 atomics into the L2-resident accumulator.
__global__ __launch_bounds__(256) void k_agg1_edges(
    const int* __restrict__ src, const int* __restrict__ dst,
    const float* __restrict__ h, const float* __restrict__ dinv,
    float* __restrict__ acc1, int E) {
  long long t = (long long)blockIdx.x * blockDim.x + threadIdx.x;
  int e = (int)(t >> 2);
  if (e >= E) return;
  int j4 = (int)(t & 3) * 4;
  int s = src[e], d = dst[e];
  float c = dinv[s] * dinv[d];
  const float4 m = *(const float4*)(h + (size_t)s * HID + j4);
  float* a = acc1 + (size_t)d * HID + j4;
  atomicAdd(a + 0, m.x * c);
  atomicAdd(a + 1, m.y * c);
  atomicAdd(a + 2, m.z * c);
  atomicAdd(a + 3, m.w * c);
}

// ---------------- ReLU + layer-2 GEMM (16->2) + out init ----------------
// h3 = relu(acc1) @ W2 ; d_out initialized with self-loop term + b2
__global__ void k_layer2_node(const float* __restrict__ acc1,
                              const float* __restrict__ W2, const float* __restrict__ b2,
                              const float* __restrict__ dinv,
                              float* __restrict__ h3, float* __restrict__ out, int N) {
  int n = blockIdx.x * blockDim.x + threadIdx.x;
  if (n >= N) return;
  float o0 = 0.0f, o1 = 0.0f;
#pragma unroll
  for (int j = 0; j < HID; ++j) {
    float v = fmaxf(acc1[(size_t)n * HID + j], 0.0f);
    o0 = __builtin_fmaf(v, W2[j * OUTF + 0], o0);
    o1 = __builtin_fmaf(v, W2[j * OUTF + 1], o1);
  }
  h3[(size_t)n * OUTF + 0] = o0;
  h3[(size_t)n * OUTF + 1] = o1;
  float dv = dinv[n];
  float c = dv * dv;
  out[(size_t)n * OUTF + 0] = o0 * c + b2[0];
  out[(size_t)n * OUTF + 1] = o1 * c + b2[1];
}

// ---------------- layer-2 edge aggregation directly into d_out ----------------
__global__ __launch_bounds__(256) void k_agg2_edges(
    const int* __restrict__ src, const int* __restrict__ dst,
    const float* __restrict__ h3, const float* __restrict__ dinv,
    float* __restrict__ out, int E) {
  int e = blockIdx.x * blockDim.x + threadIdx.x;
  if (e >= E) return;
  int s = src[e], d = dst[e];
  float c = dinv[s] * dinv[d];
  atomicAdd(&out[(size_t)d * OUTF + 0], h3[(size_t)s * OUTF + 0] * c);
  atomicAdd(&out[(size_t)d * OUTF + 1], h3[(size_t)s * OUTF + 1] * c);
}

static inline int cdiv_i(long long a, long long b) { return (int)((a + b - 1) / b); }

extern "C" void kernel_launch(void* const* d_in, const int* in_sizes, int n_in,
                              void* d_out, int out_size, void* d_ws, size_t ws_size,
                              hipStream_t stream) {
  const float* x  = (const float*)d_in[0];
  const int*   ei = (const int*)  d_in[1];
  const float* W1 = (const float*)d_in[2];
  const float* b1 = (const float*)d_in[3];
  const float* W2 = (const float*)d_in[4];
  const float* b2 = (const float*)d_in[5];
  float* out = (float*)d_out;

  const int N = in_sizes[0] / F_IN;          // 100000
  const int E = in_sizes[1] / 2;             // 3200000
  const int* src = ei;                       // edge_index[0]
  const int* dst = ei + E;                   // edge_index[1]

  float* ws   = (float*)d_ws;
  float* dinv = ws;                               // N
  float* h    = dinv + N;                         // N*16
  float* acc1 = h + (size_t)N * HID;              // N*16
  float* h3   = acc1 + (size_t)N * HID;           // N*2

  const int BS = 256;

  // degrees -> dinv
  k_deg_init <<<cdiv_i(N, BS), BS, 0, stream>>>(dinv, N);
  k_deg_count<<<cdiv_i(E, BS), BS, 0, stream>>>(dst, dinv, E);
  k_rsqrt    <<<cdiv_i(N, BS), BS, 0, stream>>>(dinv, N);

  // layer 1 GEMM (WMMA f32)
  int nTiles = cdiv_i(N, 16);
  k_gemm1_wmma<<<cdiv_i(nTiles, BS / 32), BS, 0, stream>>>(x, W1, h, N, nTiles);

  // layer 1 aggregation
  k_agg1_init <<<cdiv_i((long long)N * HID, BS), BS, 0, stream>>>(h, dinv, b1, acc1, N);
  k_agg1_edges<<<cdiv_i((long long)E * 4, BS), BS, 0, stream>>>(src, dst, h, dinv, acc1, E);

  // ReLU + layer 2 GEMM + out init (self-loop + b2 folded)
  k_layer2_node<<<cdiv_i(N, BS), BS, 0, stream>>>(acc1, W2, b2, dinv, h3, out, N);

  // layer 2 aggregation into d_out
  k_agg2_edges<<<cdiv_i(E, BS), BS, 0, stream>>>(src, dst, h3, dinv, out, E);
}